// MLE_70282844832460
// MI455X (gfx1250) — compile-verified
//
#include <hip/hip_runtime.h>

typedef __attribute__((ext_vector_type(2))) float v2f;
typedef __attribute__((ext_vector_type(4))) float v4f;
typedef __attribute__((ext_vector_type(8))) float v8f;

#define KCAND 45675
#define NB 64
#define KT ((KCAND + 15) / 16)   // 2855 k-tiles
#define CHUNKS 96
#define TPB 256

__global__ void mle_ws_init(unsigned long long* ws) {
    if (threadIdx.x < NB) ws[threadIdx.x] = 0ull;
}

// D[K-tile 16 x B-tile 16] = P_flat[K,256] x covT[256,B] (complex, Karatsuba 3-mult),
// fused per-batch argmax. Inner dim consumed in permuted 8-blocks so every lane's
// data for two WMMA steps is one contiguous b128 load (global and LDS).
__global__ __launch_bounds__(TPB) void mle_phase1(
    const float* __restrict__ cov_re, const float* __restrict__ cov_im,
    const float* __restrict__ P_re,   const float* __restrict__ P_im,
    unsigned long long* __restrict__ ws) {
    // covT staged as [n][t], row stride 260 floats (pad 4; 1040B row stride keeps
    // 16B alignment and gives each bank exactly 2 requests per wave-wide b128 read)
    __shared__ float lds_re [16][260];
    __shared__ float lds_im [16][260];
    __shared__ float lds_sum[16][260];   // re + im, for the Karatsuba product

    const int b0 = blockIdx.y * 16;
    for (int idx = threadIdx.x; idx < 16 * 256; idx += TPB) {
        int n = idx >> 8;            // batch column within tile
        int t = idx & 255;           // inner index t = i*16 + j  (covT[b][t] = cov[b][j][i])
        int i = t >> 4, j = t & 15;
        float re = cov_re[(b0 + n) * 256 + j * 16 + i];
        float im = cov_im[(b0 + n) * 256 + j * 16 + i];
        lds_re [n][t] = re;
        lds_im [n][t] = im;
        lds_sum[n][t] = re + im;
    }
    __syncthreads();

    const int lane = threadIdx.x & 31;
    const int wave = threadIdx.x >> 5;
    const int n    = lane & 15;      // A row (candidate) / B+D column (batch)
    const int h    = lane >> 4;      // half-wave selector (K-slots 2h, 2h+1)
    const int wavesPerB = CHUNKS * (TPB / 32);
    const int gw        = blockIdx.x * (TPB / 32) + wave;

    unsigned long long bestKey = 0ull;

    for (int kt = gw; kt < KT; kt += wavesPerB) {
        const int k0  = kt << 4;
        int row = k0 + n;                        // candidate row this lane loads
        row = row < KCAND ? row : (KCAND - 1);   // branchless clamp; masked at key step
        // Per 8-block: slot 2h+v <- t = kk8 + 4h + (step?2:0) + v  (same map for A and B)
        const size_t abase = (size_t)row * 256 + 4 * h;
        const float* pAre = P_re + abase;
        const float* pAim = P_im + abase;

        v8f acc1 = {};   // Re*Re
        v8f acc2 = {};   // Im*Im
        v8f acc3 = {};   // (Re+Im)*(Re+Im)
#pragma unroll 4
        for (int kk8 = 0; kk8 < 256; kk8 += 8) {
            // A fragments: one b128 covers both steps of this 8-block for this lane
            v4f are  = *(const v4f*)(pAre + kk8);
            v4f aim  = *(const v4f*)(pAim + kk8);
            v4f asum = are + aim;                 // 2x v_pk_add_f32
            // B fragments: b128 from LDS, same permuted slot->t map
            v4f bre  = *(const v4f*)&lds_re [n][kk8 + 4 * h];
            v4f bim  = *(const v4f*)&lds_im [n][kk8 + 4 * h];
            v4f bsum = *(const v4f*)&lds_sum[n][kk8 + 4 * h];
            // step 0: slots <- t = kk8 + 4h + {0,1}
            acc1 = __builtin_amdgcn_wmma_f32_16x16x4_f32(false, are.lo,  false, bre.lo,  (short)0, acc1, false, false);
            acc2 = __builtin_amdgcn_wmma_f32_16x16x4_f32(false, aim.lo,  false, bim.lo,  (short)0, acc2, false, false);
            acc3 = __builtin_amdgcn_wmma_f32_16x16x4_f32(false, asum.lo, false, bsum.lo, (short)0, acc3, false, false);
            // step 1: slots <- t = kk8 + 4h + {2,3}
            acc1 = __builtin_amdgcn_wmma_f32_16x16x4_f32(false, are.hi,  false, bre.hi,  (short)0, acc1, false, false);
            acc2 = __builtin_amdgcn_wmma_f32_16x16x4_f32(false, aim.hi,  false, bim.hi,  (short)0, acc2, false, false);
            acc3 = __builtin_amdgcn_wmma_f32_16x16x4_f32(false, asum.hi, false, bsum.hi, (short)0, acc3, false, false);
        }

        // C/D layout: lanes 0-15 VGPR v -> M=v ; lanes 16-31 -> M=v+8
        const int mBase = 8 * h;
#pragma unroll
        for (int v = 0; v < 8; v++) {
            float p1 = acc1[v], p2 = acc2[v], p3 = acc3[v];
            float re = p1 - p2;               // Karatsuba recombination
            float im = p3 - p1 - p2;
            float mag2 = fmaf(re, re, im * im);   // |z|^2: monotone in |z|, skip sqrt
            unsigned int kIdx = (unsigned int)(k0 + mBase + v);
            unsigned long long key =
                ((unsigned long long)__float_as_uint(mag2) << 32) |
                (unsigned long long)(0xFFFFFFFFu - kIdx);   // ties -> smallest k wins
            key = (kIdx < (unsigned)KCAND) ? key : 0ull;    // mask padded/clamped rows
            bestKey = key > bestKey ? key : bestKey;
        }
    }

    // lanes L and L^16 hold the same batch column n: fold, then one atomic per b
    unsigned long long other = __shfl_xor(bestKey, 16, 32);
    if (other > bestKey) bestKey = other;
    if (lane < 16) atomicMax(&ws[b0 + n], bestKey);
}

__global__ void mle_phase2(const unsigned long long* __restrict__ ws,
                           const float* __restrict__ cand_angles,
                           const float* __restrict__ cand_dist,
                           float* __restrict__ out) {
    int b = threadIdx.x;
    if (b >= NB) return;
    unsigned long long key = ws[b];
    float mag2 = __uint_as_float((unsigned int)(key >> 32));
    unsigned int k = 0xFFFFFFFFu - (unsigned int)(key & 0xFFFFFFFFull);
    bool pos = (mag2 > 0.0f) && (k < (unsigned)KCAND);
    float a0 = 0.f, a1 = 0.f, d0 = 0.f, d1 = 0.f;
    if (pos) {
        a0 = cand_angles[k * 2 + 0];
        a1 = cand_angles[k * 2 + 1];
        d0 = cand_dist[k * 2 + 0];
        d1 = cand_dist[k * 2 + 1];
    }
    out[b * 2 + 0]          = a0;   // angles_out [B,2] flat
    out[b * 2 + 1]          = a1;
    out[NB * 2 + b * 2 + 0] = d0;   // dist_out [B,2] flat, concatenated after angles
    out[NB * 2 + b * 2 + 1] = d1;
}

extern "C" void kernel_launch(void* const* d_in, const int* in_sizes, int n_in,
                              void* d_out, int out_size, void* d_ws, size_t ws_size,
                              hipStream_t stream) {
    const float* cov_re      = (const float*)d_in[0];
    const float* cov_im      = (const float*)d_in[1];
    const float* P_re        = (const float*)d_in[2];
    const float* P_im        = (const float*)d_in[3];
    const float* cand_angles = (const float*)d_in[4];
    const float* cand_dist   = (const float*)d_in[5];
    unsigned long long* ws   = (unsigned long long*)d_ws;

    mle_ws_init<<<1, 64, 0, stream>>>(ws);
    dim3 grid(CHUNKS, 4);  // x: k-strips, y: batch tiles of 16
    mle_phase1<<<grid, TPB, 0, stream>>>(cov_re, cov_im, P_re, P_im, ws);
    mle_phase2<<<1, 64, 0, stream>>>(ws, cand_angles, cand_dist, (float*)d_out);
}